// TeluguCompositeLoss_28759101014350
// MI455X (gfx1250) — compile-verified
//
#include <hip/hip_runtime.h>
#include <hip/hip_bf16.h>
#include <math.h>

// Problem constants (shapes are static per the reference).
#define BB  8
#define SS  512
#define VV  32000
#define HH  16
#define WND 20

typedef __attribute__((ext_vector_type(2))) float v2f;
typedef __attribute__((ext_vector_type(4))) float v4f;
typedef __attribute__((ext_vector_type(8))) float v8f;

// ---------------------------------------------------------------------------
// Kernel A: one block per row (b,s). Single streaming pass over V=32000 with
// online softmax: running max m, es = sum exp(x-m), tx = sum x*exp(x-m),
// sx = sum x. Emits per-row: lse = m + log(es), smooth = lse - sx/V,
// ent = lse - tx/es. Non-temporal b128 loads + global_prefetch_b8 stream.
// ---------------------------------------------------------------------------
__global__ __launch_bounds__(256) void row_stats_kernel(
    const float* __restrict__ logits, float* __restrict__ lse,
    float* __restrict__ smooth, float* __restrict__ ent) {
  const int r   = blockIdx.x;            // 0..4095
  const int tid = threadIdx.x;
  const v4f* row4 = (const v4f*)(logits + (size_t)r * VV);
  const int N4 = VV / 4;                 // 8000 float4 per row

  float m[4]  = {-INFINITY, -INFINITY, -INFINITY, -INFINITY};
  float es[4] = {0.f, 0.f, 0.f, 0.f};
  float tx[4] = {0.f, 0.f, 0.f, 0.f};
  float sx    = 0.f;

  for (int c = tid; c < N4; c += 256) {
    __builtin_prefetch((const void*)(row4 + c + 256 * 8), 0, 0); // stream ahead
    v4f x = __builtin_nontemporal_load(row4 + c);
#pragma unroll
    for (int k = 0; k < 4; ++k) {
      float xi = x[k];
      sx += xi;
      float mo = m[k];
      float mn = fmaxf(mo, xi);
      float rs = __expf(mo - mn);        // 1 when no new max; 0 on first elem
      float e  = __expf(xi - mn);
      es[k] = es[k] * rs + e;
      tx[k] = tx[k] * rs + xi * e;
      m[k]  = mn;
    }
  }
  // Merge the 4 per-component accumulators.
  float mm = m[0], ees = es[0], ttx = tx[0];
#pragma unroll
  for (int k = 1; k < 4; ++k) {
    float mn = fmaxf(mm, m[k]);
    float r1 = __expf(mm - mn), r2 = __expf(m[k] - mn);
    ees = ees * r1 + es[k] * r2;
    ttx = ttx * r1 + tx[k] * r2;
    mm  = mn;
  }

  __shared__ float shm[256], she[256], sht[256], shx[256];
  shm[tid] = mm; she[tid] = ees; sht[tid] = ttx; shx[tid] = sx;
  __syncthreads();
  for (int off = 128; off > 0; off >>= 1) {
    if (tid < off) {
      float m1 = shm[tid], m2 = shm[tid + off];
      float mn = fmaxf(m1, m2);
      float r1 = __expf(m1 - mn), r2 = __expf(m2 - mn);
      she[tid] = she[tid] * r1 + she[tid + off] * r2;
      sht[tid] = sht[tid] * r1 + sht[tid + off] * r2;
      shm[tid] = mn;
      shx[tid] += shx[tid + off];
    }
    __syncthreads();
  }
  if (tid == 0) {
    float M = shm[0], ES = she[0], T = sht[0], SX = shx[0];
    float l = M + __logf(ES);
    lse[r]    = l;
    smooth[r] = l - SX * (1.0f / VV);
    ent[r]    = l - T / ES;
  }
}

// ---------------------------------------------------------------------------
// Kernel B: one thread per row. Scattered single-element logits gathers for
// label-smoothed CE, decayed token window, consecutive-token prob, plus
// entropy accumulation. Per-block partial sums (fixed-order trees).
// ---------------------------------------------------------------------------
__global__ __launch_bounds__(256) void gather_kernel(
    const float* __restrict__ logits, const int* __restrict__ labels,
    const int* __restrict__ ids, const float* __restrict__ lse,
    const float* __restrict__ smooth, const float* __restrict__ ent,
    float* __restrict__ pB) {
  const int r = blockIdx.x * 256 + threadIdx.x;  // 0..4095
  const int b = r >> 9;
  const int s = r & (SS - 1);
  const float l = lse[r];

  // Label-smoothed CE (masked by labels != -100)
  int lab = labels[r];
  bool valid = (lab != -100);
  int lbl = valid ? lab : 0;
  float nll = l - logits[(size_t)r * VV + lbl];
  float ce = valid ? (0.9f * nll + 0.1f * smooth[r]) : 0.f;
  float vc = valid ? 1.f : 0.f;

  // Decayed token window
  float tok_acc = 0.f;
  for (int d = 0; d < WND; ++d) {
    int pos = s - 1 - d;
    if (pos < 0) break;                      // pos decreases monotonically
    int tok = ids[b * SS + pos];
    if (tok != 0) {
      float p = __expf(logits[(size_t)r * VV + tok] - l);
      tok_acc += p * __expf(-(float)d * 0.2f);
    }
  }

  // Consecutive-token probability: probs[b, s, input_ids[b, s-1]]
  float cs = 0.f, cc = 0.f;
  if (s >= 1) {
    int prev = ids[b * SS + s - 1];
    if (prev != 0) {
      cs = __expf(logits[(size_t)r * VV + prev] - l);
      cc = 1.f;
    }
  }

  float e  = ent[r];
  float dv = fmaxf(0.f, 2.f - e);

  __shared__ float sh[256];
  float vals[7] = {ce, vc, tok_acc, cs, cc, e, dv};
  for (int k = 0; k < 7; ++k) {
    sh[threadIdx.x] = vals[k];
    __syncthreads();
    for (int off = 128; off > 0; off >>= 1) {
      if (threadIdx.x < off) sh[threadIdx.x] += sh[threadIdx.x + off];
      __syncthreads();
    }
    if (threadIdx.x == 0) pB[blockIdx.x * 7 + k] = sh[0];
    __syncthreads();
  }
}

// ---------------------------------------------------------------------------
// Kernel C: one block per batch. Exact trigram n-gram repetition term.
// Phase 1: first-occurrence flag per position. Phase 2: all (i,j) pairs with
// 2-token prefix match; gather probs[b, i+2, ids[j+2]].
// ---------------------------------------------------------------------------
__global__ __launch_bounds__(256) void ngram_kernel(
    const float* __restrict__ logits, const int* __restrict__ ids_g,
    const float* __restrict__ lse, float* __restrict__ pC) {
  const int b = blockIdx.x;
  const int tid = threadIdx.x;
  const int M = SS - 2;                    // 510
  __shared__ int ids[SS];
  __shared__ int first[SS];

  for (int i = tid; i < SS; i += 256) ids[i] = ids_g[b * SS + i];
  __syncthreads();

  for (int j = tid; j < M; j += 256) {
    int a0 = ids[j], a1 = ids[j + 1], a2 = ids[j + 2];
    int f = 1;
    for (int jp = 0; jp < j; ++jp) {
      if (ids[jp] == a0 && ids[jp + 1] == a1 && ids[jp + 2] == a2) { f = 0; break; }
    }
    first[j] = f;
  }
  __syncthreads();

  float sum = 0.f, cnt = 0.f;
  for (int idx = tid; idx < M * M; idx += 256) {
    int i = idx / M;
    int j = idx - i * M;
    if (ids[i] == ids[j] && ids[i + 1] == ids[j + 1]) {  // prefix equal
      int c = ids[j + 2];
      if (first[j] && c != 0) {
        cnt += 1.f;
        int row = b * SS + i + 2;
        sum += __expf(logits[(size_t)row * VV + c] - lse[row]);
      }
    }
  }

  __shared__ float shs[256], shc[256];
  shs[tid] = sum; shc[tid] = cnt;
  __syncthreads();
  for (int off = 128; off > 0; off >>= 1) {
    if (tid < off) { shs[tid] += shs[tid + off]; shc[tid] += shc[tid + off]; }
    __syncthreads();
  }
  if (tid == 0) { pC[b * 2] = shs[0]; pC[b * 2 + 1] = shc[0]; }
}

// ---------------------------------------------------------------------------
// Kernel D: coverage term via v_wmma_f32_16x16x4_f32 (head-mean = ones-row
// GEMM), then final scalar combine. A = 1/16 uniform over K, so the result is
// invariant to K-slot permutation in B's layout; only N = lane%16 matters.
// Each wave handles 16 attention columns; 4 WMMAs accumulate the 16 heads.
// ---------------------------------------------------------------------------
__global__ __launch_bounds__(256) void final_kernel(
    const float* __restrict__ attn, const float* __restrict__ coverage,
    const float* __restrict__ pB, const float* __restrict__ pC,
    float* __restrict__ out) {
  const int tid  = threadIdx.x;
  const int wave = tid >> 5;               // 8 waves
  const int lane = tid & 31;
  const int n     = lane & 15;             // column within 16-wide tile
  const int khalf = (lane >> 4) << 1;      // K-slots {0,1} / {2,3} per half-wave

  float covAcc = 0.f;
  v2f a = {0.0625f, 0.0625f};              // A[m,k] = 1/16 for all m,k
  for (int t = wave; t < BB * (SS / 16); t += 8) {   // 256 tiles total
    int b  = t >> 5;
    int s0 = (t & 31) << 4;
    v8f c = {0.f, 0.f, 0.f, 0.f, 0.f, 0.f, 0.f, 0.f};
#pragma unroll
    for (int k0 = 0; k0 < HH; k0 += 4) {
      size_t base = (((size_t)(b * HH + k0 + khalf) * SS) + (SS - 1)) * SS + (s0 + n);
      v2f bv = {attn[base], attn[base + (size_t)SS * SS]};   // two heads/lane
      c = __builtin_amdgcn_wmma_f32_16x16x4_f32(false, a, false, bv,
                                                (short)0, c, false, false);
    }
    if (lane < 16)                         // D[0][n] on lanes 0..15
      covAcc += fminf(c[0], coverage[b * SS + s0 + n]);
  }

  __shared__ float sh[256];
  sh[tid] = covAcc;
  __syncthreads();
  for (int off = 128; off > 0; off >>= 1) {
    if (tid < off) sh[tid] += sh[tid + off];
    __syncthreads();
  }

  if (tid == 0) {
    float covSum = sh[0];
    float ce = 0, vc = 0, tok = 0, cs = 0, cc = 0, ents = 0, divs = 0;
    for (int i = 0; i < 16; ++i) {
      ce   += pB[i * 7 + 0];
      vc   += pB[i * 7 + 1];
      tok  += pB[i * 7 + 2];
      cs   += pB[i * 7 + 3];
      cc   += pB[i * 7 + 4];
      ents += pB[i * 7 + 5];
      divs += pB[i * 7 + 6];
    }
    float ng = 0, nc = 0;
    for (int b = 0; b < BB; ++b) { ng += pC[b * 2]; nc += pC[b * 2 + 1]; }

    float lm         = ce / fmaxf(vc, 1.f);
    float token_loss = tok / (float)(BB * (SS - 1));
    float consec     = cs / fmaxf(cc, 1.f);
    float ngram      = ng / fmaxf(nc, 1.f);
    float anti_rep   = 0.3f * 0.3f * token_loss + 0.8f * 0.3f * consec +
                       0.5f * 0.3f * ngram;
    float prosody    = 0.3f * 0.1f * (ents / (float)(BB * SS));
    float diversity  = 0.1f * (divs / (float)(BB * SS));
    float cov_loss   = 0.2f * covSum / (float)BB;
    out[0] = lm + anti_rep + prosody + diversity + cov_loss;
  }
}

// ---------------------------------------------------------------------------
extern "C" void kernel_launch(void* const* d_in, const int* in_sizes, int n_in,
                              void* d_out, int out_size, void* d_ws, size_t ws_size,
                              hipStream_t stream) {
  const float* logits = (const float*)d_in[0];   // (8,512,32000) f32
  const int*   labels = (const int*)d_in[1];     // (8,512) i32
  const int*   ids    = (const int*)d_in[2];     // (8,512) i32
  const float* attn   = (const float*)d_in[3];   // (8,16,512,512) f32
  const float* cover  = (const float*)d_in[4];   // (8,512) f32
  float* out = (float*)d_out;

  float* ws     = (float*)d_ws;
  float* lse    = ws;             // 4096
  float* smooth = ws + 4096;      // 4096
  float* ent    = ws + 8192;      // 4096
  float* pB     = ws + 12288;     // 16 blocks * 7
  float* pC     = ws + 12416;     // 8 batches * 2

  hipLaunchKernelGGL(row_stats_kernel, dim3(BB * SS), dim3(256), 0, stream,
                     logits, lse, smooth, ent);
  hipLaunchKernelGGL(gather_kernel, dim3(16), dim3(256), 0, stream,
                     logits, labels, ids, lse, smooth, ent, pB);
  hipLaunchKernelGGL(ngram_kernel, dim3(BB), dim3(256), 0, stream,
                     logits, ids, lse, pC);
  hipLaunchKernelGGL(final_kernel, dim3(1), dim3(256), 0, stream,
                     attn, cover, pB, pC, out);
}